// WassersteinBottleneck_12386685682169
// MI455X (gfx1250) — compile-verified
//
#include <hip/hip_runtime.h>
#include <hip/hip_bf16.h>

typedef float v2f __attribute__((ext_vector_type(2)));
typedef float v8f __attribute__((ext_vector_type(8)));

#define NROWS 8192   // B*T = 4*2048
#define CDIM  64

// ---------------------------------------------------------------------------
// Kernel 1: transpose latents [4, 64, 2048] -> x row-major [8192, 64] in ws.
// Also zero the three f64 accumulators (stream order guarantees visibility).
// ---------------------------------------------------------------------------
__global__ void mmd_transpose_init(const float* __restrict__ latents,
                                   float* __restrict__ xbuf,
                                   double* __restrict__ accum) {
    int idx = blockIdx.x * blockDim.x + threadIdx.x;   // 0 .. 8192*64-1
    int c   = idx & 63;          // channel
    int row = idx >> 6;          // b*2048 + t
    int b   = row >> 11;
    int t   = row & 2047;
    xbuf[idx] = latents[((b << 6) + c) * 2048 + t];
    if (idx < 3) accum[idx] = 0.0;
}

// ---------------------------------------------------------------------------
// Kernel 2: row squared-norms for x (from ws) and y (noise). blockIdx.y picks.
// ---------------------------------------------------------------------------
__global__ void mmd_rownorm(const float* __restrict__ xbuf,
                            const float* __restrict__ noise,
                            float* __restrict__ x2,
                            float* __restrict__ y2) {
    int r = blockIdx.x * blockDim.x + threadIdx.x;     // row 0..8191
    const float4* src = (const float4*)((blockIdx.y == 0) ? xbuf : noise);
    float*        dst = (blockIdx.y == 0) ? x2 : y2;
    const float4* p = src + r * (CDIM / 4);
    float s = 0.f;
#pragma unroll
    for (int i = 0; i < CDIM / 4; ++i) {
        float4 v = p[i];
        s += v.x * v.x + v.y * v.y + v.z * v.z + v.w * v.w;
    }
    dst[r] = s;
}

// ---------------------------------------------------------------------------
// Kernel 3: WMMA Gram + exp + reduce. One wave -> 32x64 output patch,
// 2x4 grid of v_wmma_f32_16x16x4_f32 accumulators, K = 64 in 16 steps.
// Pair p: 0=(x,x) 1=(y,y) 2=(x,y). Sum of exp values -> accum[p] (f64 atomic).
// ---------------------------------------------------------------------------
__global__ __launch_bounds__(256) void mmd_wmma(
        const float* __restrict__ xbuf, const float* __restrict__ noise,
        const float* __restrict__ x2,   const float* __restrict__ y2,
        double* __restrict__ accum) {
    const int lane   = threadIdx.x & 31;
    const int waveId = blockIdx.x * 8 + (threadIdx.x >> 5);
    const int p   = waveId >> 15;        // / (256*128)
    const int rem = waveId & 32767;
    const int m0  = (rem >> 7) << 5;     // mBlock*32
    const int n0  = (rem & 127) << 6;    // nBlock*64

    const float* A  = (p == 1) ? noise : xbuf;
    const float* Bm = (p == 0) ? xbuf  : noise;
    const float* a2 = (p == 1) ? y2 : x2;
    const float* b2 = (p == 0) ? x2 : y2;

    const int r  = lane & 15;   // M/N row within 16-tile
    const int kh = lane >> 4;   // selects K pair {0,1} vs {2,3}

    v8f acc[2][4];
#pragma unroll
    for (int mi = 0; mi < 2; ++mi)
#pragma unroll
        for (int ni = 0; ni < 4; ++ni) acc[mi][ni] = v8f{};

    // ISA 16x4 f32 A layout: lane L holds A[M=L%16, K=2*(L/16) + {0,1}] -> float2.
    // B = X_B^T of a row-major matrix => identical per-lane formula on Bm.
    const float* aBase = A  + (m0 + r) * CDIM + (kh << 1);
    const float* bBase = Bm + (n0 + r) * CDIM + (kh << 1);

#pragma unroll
    for (int k = 0; k < CDIM; k += 4) {
        v2f aF[2], bF[4];
#pragma unroll
        for (int mi = 0; mi < 2; ++mi)
            aF[mi] = *(const v2f*)(aBase + (mi * 16) * CDIM + k);
#pragma unroll
        for (int ni = 0; ni < 4; ++ni)
            bF[ni] = *(const v2f*)(bBase + (ni * 16) * CDIM + k);
#pragma unroll
        for (int mi = 0; mi < 2; ++mi)
#pragma unroll
            for (int ni = 0; ni < 4; ++ni)
                acc[mi][ni] = __builtin_amdgcn_wmma_f32_16x16x4_f32(
                    false, aF[mi], false, bF[ni],
                    (short)0, acc[mi][ni], false, false);
    }

    // exp(-sq/4096) = exp2(sq * cexp), sq = a2[m] + b2[n] - 2*G
    const float cexp = -1.4426950408889634f / 4096.0f;
    float b2v[4];
#pragma unroll
    for (int ni = 0; ni < 4; ++ni) b2v[ni] = b2[n0 + ni * 16 + r];

    float wsum = 0.f;
#pragma unroll
    for (int mi = 0; mi < 2; ++mi) {
#pragma unroll
        for (int e = 0; e < 8; ++e) {
            // C/D layout: VGPR e, lanes 0-15: M=e ; lanes 16-31: M=e+8 ; N=lane%16
            float a2v = a2[m0 + mi * 16 + kh * 8 + e];
#pragma unroll
            for (int ni = 0; ni < 4; ++ni) {
                float g  = acc[mi][ni][e];
                float sq = a2v + b2v[ni] - 2.0f * g;
                wsum += exp2f(sq * cexp);
            }
        }
    }

    // wave32 reduction
#pragma unroll
    for (int off = 16; off > 0; off >>= 1)
        wsum += __shfl_xor(wsum, off, 32);
    if (lane == 0) atomicAdd(&accum[p], (double)wsum);
}

// ---------------------------------------------------------------------------
// Kernel 4: combine to scalar MMD.
// ---------------------------------------------------------------------------
__global__ void mmd_finalize(const double* __restrict__ accum,
                             float* __restrict__ out) {
    if (threadIdx.x == 0 && blockIdx.x == 0) {
        const double inv = 1.0 / ((double)NROWS * (double)NROWS);
        out[0] = (float)((accum[0] + accum[1] - 2.0 * accum[2]) * inv);
    }
}

// ---------------------------------------------------------------------------
extern "C" void kernel_launch(void* const* d_in, const int* in_sizes, int n_in,
                              void* d_out, int out_size, void* d_ws, size_t ws_size,
                              hipStream_t stream) {
    const float* latents = (const float*)d_in[0];  // [4, 64, 2048] f32
    const float* noise   = (const float*)d_in[1];  // [8192, 64]   f32

    float*  xbuf  = (float*)d_ws;                  // 8192*64 f32 = 2 MB
    float*  x2    = xbuf + NROWS * CDIM;           // 8192 f32
    float*  y2    = x2 + NROWS;                    // 8192 f32
    double* accum = (double*)(y2 + NROWS);         // 3 f64 (8B aligned)
    float*  out   = (float*)d_out;

    mmd_transpose_init<<<(NROWS * CDIM) / 256, 256, 0, stream>>>(latents, xbuf, accum);
    mmd_rownorm<<<dim3(NROWS / 256, 2), 256, 0, stream>>>(xbuf, noise, x2, y2);
    // 3 pairs * (8192/32) mBlocks * (8192/64) nBlocks = 98304 waves, 8 waves/block
    mmd_wmma<<<98304 / 8, 256, 0, stream>>>(xbuf, noise, x2, y2, accum);
    mmd_finalize<<<1, 32, 0, stream>>>(accum, out);
}